// TiaRaGNNModel_46858093199625
// MI455X (gfx1250) — compile-verified
//
#include <hip/hip_runtime.h>

#define NN    100000
#define EE    1600000
#define F_INN 128
#define HID   64
#define TOUT  10
#define ALPHAC 0.1f

typedef float v2f __attribute__((ext_vector_type(2)));
typedef float v8f __attribute__((ext_vector_type(8)));

// ---------------- utility fills ----------------
__global__ void fill_f32(float* __restrict__ p, float v, int n) {
    int i = blockIdx.x * blockDim.x + threadIdx.x;
    if (i < n) p[i] = v;
}

// ---------------- max(edge_time) ----------------
__global__ void max_reduce(const float* __restrict__ t, int n, unsigned int* __restrict__ out) {
    __shared__ float s[256];
    float m = 0.0f;                       // edge_time is uniform [0,1): nonneg
    for (int i = blockIdx.x * blockDim.x + threadIdx.x; i < n; i += gridDim.x * blockDim.x)
        m = fmaxf(m, t[i]);
    s[threadIdx.x] = m;
    __syncthreads();
    for (int w = 128; w > 0; w >>= 1) {
        if (threadIdx.x < w) s[threadIdx.x] = fmaxf(s[threadIdx.x], s[threadIdx.x + w]);
        __syncthreads();
    }
    if (threadIdx.x == 0) atomicMax(out, __float_as_uint(s[0])); // bit-compare OK for nonneg floats
}

// ---------------- decay + out-degree ----------------
__global__ void decay_deg(const float* __restrict__ et, const long long* __restrict__ ei,
                          const unsigned int* __restrict__ tmax_bits,
                          float* __restrict__ decay, float* __restrict__ deg) {
    int e = blockIdx.x * blockDim.x + threadIdx.x;
    if (e >= EE) return;
    float tmax = __uint_as_float(*tmax_bits);
    float d = expf(-ALPHAC * (tmax - et[e]));
    decay[e] = d;
    atomicAdd(&deg[(int)ei[e]], d);       // row = ei[0][e]
}

__global__ void inv_sqrt_pos(float* __restrict__ p, int n) {
    int i = blockIdx.x * blockDim.x + threadIdx.x;
    if (i >= n) return;
    float d = p[i];
    p[i] = (d > 0.0f) ? rsqrtf(d) : 0.0f;
}

// ew = dinv[row]*decay*dinv[col]; deg2[col] += ew (deg2 pre-filled with 1.0 for self loops)
__global__ void ew_deg2(const long long* __restrict__ ei, const float* __restrict__ dinv,
                        float* __restrict__ bufE, float* __restrict__ deg2) {
    int e = blockIdx.x * blockDim.x + threadIdx.x;
    if (e >= EE) return;
    int r = (int)ei[e], c = (int)ei[EE + e];
    float w = bufE[e] * dinv[r] * dinv[c];
    bufE[e] = w;
    atomicAdd(&deg2[c], w);
}

// norm = dinv2[row]*ew*dinv2[col]
__global__ void norm_k(const long long* __restrict__ ei, const float* __restrict__ dinv2,
                       float* __restrict__ bufE) {
    int e = blockIdx.x * blockDim.x + threadIdx.x;
    if (e >= EE) return;
    int r = (int)ei[e], c = (int)ei[EE + e];
    bufE[e] = bufE[e] * dinv2[r] * dinv2[c];
}

// ---------------- fp32 WMMA GEMM: D[NN x 64] = A[NN x K] @ B[K x 64] ----------------
__global__ __launch_bounds__(256) void gemm_wmma(const float* __restrict__ A,
                                                 const float* __restrict__ B,
                                                 float* __restrict__ D, int K) {
    const int lane = threadIdx.x & 31;
    const int wave = threadIdx.x >> 5;
    const int tile = blockIdx.x * 8 + wave;         // 4 col-tiles per row-tile
    const int m0 = (tile >> 2) * 16;
    const int n0 = (tile & 3) * 16;
    if (m0 >= NN) return;                           // uniform per wave (EXEC stays all-1)
    const int l15  = lane & 15;
    const int koff = (lane < 16) ? 0 : 2;
    const float* arow = A + (size_t)(m0 + l15) * K + koff;
    const float* bcol = B + (size_t)koff * HID + (n0 + l15);
    v8f c = {};
    for (int k = 0; k < K; k += 4) {
        v2f a, b;
        a.x = arow[k];
        a.y = arow[k + 1];
        b.x = bcol[(size_t)k * HID];
        b.y = bcol[(size_t)(k + 1) * HID];
        c = __builtin_amdgcn_wmma_f32_16x16x4_f32(false, a, false, b, (short)0, c, false, false);
    }
    const int drow = m0 + ((lane < 16) ? 0 : 8);
    float* dptr = D + (size_t)drow * HID + (n0 + l15);
#pragma unroll
    for (int r = 0; r < 8; ++r) dptr[(size_t)r * HID] = c[r];
}

// agg[n][k] = dinv2[n]^2 * h[n][k]   (self-loop term: norm_loop = dinv2[n]*1*dinv2[n])
__global__ void agg_init(const float* __restrict__ h, const float* __restrict__ dinv2,
                         float* __restrict__ agg) {
    int i = blockIdx.x * blockDim.x + threadIdx.x;
    if (i >= NN * HID) return;
    float s = dinv2[i >> 6];
    agg[i] = s * s * h[i];
}

// agg[col][k] += norm[e] * h[row][k]; 64 consecutive threads share one edge -> coalesced
__global__ void scatter_k(const long long* __restrict__ ei, const float* __restrict__ nrm,
                          const float* __restrict__ h, float* __restrict__ agg) {
    int tid = blockIdx.x * blockDim.x + threadIdx.x;   // EE*HID = 102.4M fits in int
    if (tid >= EE * HID) return;
    int e = tid >> 6, k = tid & 63;
    int r = (int)ei[e], c = (int)ei[EE + e];
    atomicAdd(&agg[c * HID + k], nrm[e] * h[r * HID + k]);
}

__global__ void relu_bias(float* __restrict__ agg, const float* __restrict__ b) {
    int i = blockIdx.x * blockDim.x + threadIdx.x;
    if (i >= NN * HID) return;
    float v = agg[i] + b[i & 63];
    agg[i] = v > 0.0f ? v : 0.0f;
}

// out[n][t] = h[n][:] @ Wout[:,t] + bout[t]   (64x10 readout, tiny)
__global__ void readout(const float* __restrict__ h, const float* __restrict__ Wout,
                        const float* __restrict__ bout, float* __restrict__ out) {
    int i = blockIdx.x * blockDim.x + threadIdx.x;
    if (i >= NN * TOUT) return;
    int n = i / TOUT, t = i - n * TOUT;
    float acc = bout[t];
#pragma unroll
    for (int k = 0; k < HID; ++k) acc = fmaf(h[n * HID + k], Wout[k * TOUT + t], acc);
    out[i] = acc;
}

extern "C" void kernel_launch(void* const* d_in, const int* in_sizes, int n_in,
                              void* d_out, int out_size, void* d_ws, size_t ws_size,
                              hipStream_t stream) {
    const float*     x    = (const float*)d_in[0];
    const long long* ei   = (const long long*)d_in[1];   // int64 [2,E]
    const float*     et   = (const float*)d_in[2];
    const float*     W1   = (const float*)d_in[3];
    const float*     b1   = (const float*)d_in[4];
    const float*     W2   = (const float*)d_in[5];
    const float*     b2   = (const float*)d_in[6];
    const float*     Wout = (const float*)d_in[7];
    const float*     bout = (const float*)d_in[8];
    float*           out  = (float*)d_out;

    char* ws = (char*)d_ws;
    size_t off = 0;
    auto alloc = [&](size_t bytes) {
        void* p = ws + off;
        off = (off + bytes + 255) & ~(size_t)255;
        return p;
    };
    float*        bufE  = (float*)alloc(sizeof(float) * EE);        // decay -> ew -> norm
    float*        dinvA = (float*)alloc(sizeof(float) * NN);        // deg  -> dinv
    float*        dinvB = (float*)alloc(sizeof(float) * NN);        // deg2 -> dinv2
    float*        bufH1 = (float*)alloc(sizeof(float) * NN * HID);  // h (gemm output)
    float*        bufH2 = (float*)alloc(sizeof(float) * NN * HID);  // agg / relu
    unsigned int* tmaxb = (unsigned int*)alloc(sizeof(unsigned int));

    const int TB = 256;
    const int gE  = (EE + TB - 1) / TB;
    const int gN  = (NN + TB - 1) / TB;
    const int gNH = (NN * HID + TB - 1) / TB;
    const int gEH = (EE * HID) / TB;          // exact: 400000
    const int gGEMM = (NN / 16) * 4 / 8;      // 25000 tiles / 8 waves = 3125

    // init
    fill_f32<<<1, 32, 0, stream>>>((float*)tmaxb, 0.0f, 1);
    fill_f32<<<gN, TB, 0, stream>>>(dinvA, 0.0f, NN);
    fill_f32<<<gN, TB, 0, stream>>>(dinvB, 1.0f, NN);   // self-loop weight baked in

    // edge-weight pipeline
    max_reduce<<<1024, TB, 0, stream>>>(et, EE, tmaxb);
    decay_deg<<<gE, TB, 0, stream>>>(et, ei, tmaxb, bufE, dinvA);
    inv_sqrt_pos<<<gN, TB, 0, stream>>>(dinvA, NN);
    ew_deg2<<<gE, TB, 0, stream>>>(ei, dinvA, bufE, dinvB);
    inv_sqrt_pos<<<gN, TB, 0, stream>>>(dinvB, NN);
    norm_k<<<gE, TB, 0, stream>>>(ei, dinvB, bufE);

    // GCN layer 1: h1 = x @ W1 ; agg = norm-weighted aggregate ; relu(+b1)
    gemm_wmma<<<gGEMM, 256, 0, stream>>>(x, W1, bufH1, F_INN);
    agg_init<<<gNH, TB, 0, stream>>>(bufH1, dinvB, bufH2);
    scatter_k<<<gEH, TB, 0, stream>>>(ei, bufE, bufH1, bufH2);
    relu_bias<<<gNH, TB, 0, stream>>>(bufH2, b1);

    // GCN layer 2
    gemm_wmma<<<gGEMM, 256, 0, stream>>>(bufH2, W2, bufH1, HID);
    agg_init<<<gNH, TB, 0, stream>>>(bufH1, dinvB, bufH2);
    scatter_k<<<gEH, TB, 0, stream>>>(ei, bufE, bufH1, bufH2);
    relu_bias<<<gNH, TB, 0, stream>>>(bufH2, b2);

    // readout
    readout<<<(NN * TOUT + TB - 1) / TB, TB, 0, stream>>>(bufH2, Wout, bout, out);
}